// network_76312978915751
// MI455X (gfx1250) — compile-verified
//
#include <hip/hip_runtime.h>

// ---------------- problem constants (reference: B=16, H=W=1024) -------------
#define HH 1024
#define WW 1024
#define TILE 32

// LDS region geometry (coords relative to the 32x32 output tile origin)
#define UP 46            // u pitch;   u region rows/cols [-6, 38)  -> 44x44
#define UR0 (-6)
#define KP 42            // k1 pitch;  k1 region        [-4, 36)  -> 40x40
#define KR0 (-4)
#define GP 38            // grad pitch; grad region      [-2, 34)  -> 36x36
#define GR0 (-2)

typedef float v2f __attribute__((ext_vector_type(2)));
typedef float v8f __attribute__((ext_vector_type(8)));

typedef __attribute__((address_space(1))) int gint_t;   // global int
typedef __attribute__((address_space(3))) int lint_t;   // LDS int

// CDNA5 async global->LDS path (ASYNCcnt-tracked). Guarded so that if this
// toolchain doesn't declare the builtins we fall back to the
// synchronous load loop.
#if __has_builtin(__builtin_amdgcn_global_load_async_to_lds_b32) && \
    __has_builtin(__builtin_amdgcn_s_wait_asynccnt)
#define USE_ASYNC_LDS 1
#endif

// ---------------------------------------------------------------------------
// 16x16 output tile of a 5x5 correlation via banded-matmul WMMA:
//   y[r,c] = sum_dr sum_{p=0..19} src(ro+r+dr-2, co+p-2) * Wb_dr[p,c]
// K chunked 20 -> 5 chunks of 4; one v_wmma_f32_16x16x4_f32 per (dr,chunk).
// A operand (image): lane m=lane&15 rows, kk=2*(lane>>4) K-halves.
// B operand (banded weights) precomputed in LDS: layout [dr][j][lane][2].
// c00 is always even and pitches are even, so v2f LDS reads stay 8B aligned.
// ---------------------------------------------------------------------------
__device__ __forceinline__ v8f conv16(const float* __restrict__ src, int pitch,
                                      int r00, int c00,
                                      const float* __restrict__ Bset,
                                      int lane, v8f acc)
{
    const int m  = lane & 15;
    const int kk = (lane >> 4) << 1;
    int abase = (r00 + m) * pitch + c00 + kk;
    const float* bb = Bset + lane * 2;
#pragma unroll
    for (int dr = 0; dr < 5; ++dr) {
#pragma unroll
        for (int j = 0; j < 5; ++j) {
            v2f a = *(const v2f*)(src + abase + 4 * j);
            v2f b = *(const v2f*)(bb + (dr * 5 + j) * 64);
            acc = __builtin_amdgcn_wmma_f32_16x16x4_f32(
                false, a, false, b, (short)0, acc, false, false);
        }
        abase += pitch;
    }
    return acc;
}

// Scatter a 16x16 f32 C/D tile (8 VGPRs) to LDS per the CDNA5 layout:
// VGPR v: lanes 0-15 -> M=v, lanes 16-31 -> M=v+8; N = lane&15.
__device__ __forceinline__ void store_tile(float* dst, int pitch, int r0, int c0,
                                           int lane, v8f d, bool do_relu)
{
    const int n  = lane & 15;
    const int kh = lane >> 4;
#pragma unroll
    for (int v = 0; v < 8; ++v) {
        float x = d[v];
        x = do_relu ? fmaxf(x, 0.0f) : x;
        dst[(r0 + v + 8 * kh) * pitch + (c0 + n)] = x;
    }
}

// Build banded B matrices into LDS: B[set][dr][j][lane][2]
//   B_j[k,c] = w(set,dr, 4j+k-c) when 0 <= 4j+k-c <= 4 else 0
template <typename F>
__device__ __forceinline__ void buildB(float* B_lds, int nsets, int tid, F wf)
{
    for (int p = tid; p < nsets * 25 * 32; p += 256) {
        int lane = p & 31;
        int q    = p >> 5;            // set*25 + dr*5 + j
        int j    = q % 5;
        int dr   = (q / 5) % 5;
        int set  = q / 25;
        int c    = lane & 15;
        int kk   = (lane >> 4) << 1;
        int s0   = 4 * j + kk - c;
        int s1   = s0 + 1;
        float w0 = (s0 >= 0 && s0 <= 4) ? wf(set, dr, s0) : 0.0f;
        float w1 = (s1 >= 0 && s1 <= 4) ? wf(set, dr, s1) : 0.0f;
        B_lds[p * 2 + 0] = w0;
        B_lds[p * 2 + 1] = w1;
    }
}

// ---------------------------------------------------------------------------
// One explicit-Euler layer:
//   k1   = relu(conv5(U, K1))                (2 channels)
//   grad = sum_ch conv5(K2[ch]*k1[ch], flip(K1[ch]))
//   Unew = U + (dt/3) * conv5(grad, A)
// Fully fused per 32x32 tile with circular halo loaded once from global.
// ---------------------------------------------------------------------------
__global__ __launch_bounds__(256) void rd_layer(
    const float* __restrict__ Uin, float* __restrict__ Uout,
    const float* __restrict__ K1w, const float* __restrict__ K2w,
    const float* __restrict__ Aww)
{
    __shared__ __align__(16) float u_lds[44 * UP];
    __shared__ __align__(16) float k1_lds[2 * 40 * KP];
    __shared__ __align__(16) float g_lds[36 * GP];
    __shared__ __align__(16) float B_lds[2 * 25 * 64];

    const int tid  = threadIdx.x;
    const int lane = tid & 31;
    const int wv   = tid >> 5;
    const int tr   = blockIdx.y * TILE;
    const int tc   = blockIdx.x * TILE;
    const long long bofs = (long long)blockIdx.z * (long long)(HH * WW);
    const float* Ub = Uin + bofs;
    float*       Ob = Uout + bofs;

    __builtin_prefetch(Ub + (long long)tr * WW + tc, 0, 1);   // global_prefetch_b8

    // ---- 44x44 halo tile with circular wrap: global -> LDS ---------------
#if defined(USE_ASYNC_LDS)
    // Direct async copy (global_load_async_to_lds_b32, ASYNCcnt-tracked);
    // overlaps with the banded-B build below.
    for (int i = tid; i < 44 * 44; i += 256) {
        int lr = i / 44, lc = i - lr * 44;
        int rg = tr + lr - 6;
        rg += (rg < 0) ? HH : 0;  rg -= (rg >= HH) ? HH : 0;
        int cg = tc + lc - 6;
        cg += (cg < 0) ? WW : 0;  cg -= (cg >= WW) ? WW : 0;
        __builtin_amdgcn_global_load_async_to_lds_b32(
            (gint_t*)(Ub + rg * WW + cg),
            (lint_t*)&u_lds[lr * UP + lc],
            0, 0);
    }
#else
    for (int i = tid; i < 44 * 44; i += 256) {
        int lr = i / 44, lc = i - lr * 44;
        int rg = tr + lr - 6;
        rg += (rg < 0) ? HH : 0;  rg -= (rg >= HH) ? HH : 0;
        int cg = tc + lc - 6;
        cg += (cg < 0) ? WW : 0;  cg -= (cg >= WW) ? WW : 0;
        u_lds[lr * UP + lc] = Ub[rg * WW + cg];
    }
#endif

    // banded weights for stage 1: K1 (2 output channels) -- overlaps the
    // async halo copy above.
    buildB(B_lds, 2, tid,
           [&](int s, int dr, int dc) { return K1w[s * 25 + dr * 5 + dc]; });

#if defined(USE_ASYNC_LDS)
    __builtin_amdgcn_s_wait_asynccnt(0);   // my async copies are in LDS
#endif
    __syncthreads();

    // ---- stage 1: k1 = relu(conv(u, K1)) on [-4,36)^2 --------------------
    {
        const int O1[3] = {-4, 12, 20};          // overlapping 16-tiles cover exactly
        for (int job = wv; job < 18; job += 8) { // 3x3 positions x 2 channels
            int ch = job / 9, t = job - ch * 9;
            int oy = O1[t / 3], ox = O1[t % 3];
            v8f acc = {0, 0, 0, 0, 0, 0, 0, 0};
            acc = conv16(u_lds, UP, oy - 2 - UR0, ox - 2 - UR0,
                         B_lds + ch * 1600, lane, acc);
            store_tile(k1_lds + ch * 40 * KP, KP, oy - KR0, ox - KR0, lane, acc, true);
        }
    }
    __syncthreads();
    // banded weights for stage 2: K2[ch] * flip(K1[ch])  (transposed conv)
    buildB(B_lds, 2, tid, [&](int s, int dr, int dc) {
        return K2w[s] * K1w[s * 25 + (4 - dr) * 5 + (4 - dc)];
    });
    __syncthreads();

    // ---- stage 2: grad = sum_ch conv(k1, wT) on [-2,34)^2 ----------------
    {
        const int O2[3] = {-2, 14, 18};
        for (int job = wv; job < 9; job += 8) {
            int oy = O2[job / 3], ox = O2[job % 3];
            v8f acc = {0, 0, 0, 0, 0, 0, 0, 0};
            acc = conv16(k1_lds,            KP, oy - 2 - KR0, ox - 2 - KR0,
                         B_lds,        lane, acc);
            acc = conv16(k1_lds + 40 * KP,  KP, oy - 2 - KR0, ox - 2 - KR0,
                         B_lds + 1600, lane, acc);
            store_tile(g_lds, GP, oy - GR0, ox - GR0, lane, acc, false);
        }
    }
    __syncthreads();
    buildB(B_lds, 1, tid,
           [&](int s, int dr, int dc) { (void)s; return Aww[dr * 5 + dc]; });
    __syncthreads();

    // ---- stage 3: Unew = U + (dt/3) * conv(grad, A) on [0,32)^2 ----------
    {
        const float coef = 0.1f / 3.0f;
        for (int job = wv; job < 4; job += 8) {
            int oy = (job >> 1) * 16, ox = (job & 1) * 16;
            v8f acc = {0, 0, 0, 0, 0, 0, 0, 0};
            acc = conv16(g_lds, GP, oy - 2 - GR0, ox - 2 - GR0, B_lds, lane, acc);
            const int n = lane & 15, kh = lane >> 4;
#pragma unroll
            for (int v = 0; v < 8; ++v) {
                int m = oy + v + 8 * kh;
                int c = ox + n;
                float uv = u_lds[(m - UR0) * UP + (c - UR0)];
                Ob[(tr + m) * WW + (tc + c)] = uv + coef * acc[v];
            }
        }
    }
}

// ---------------------------------------------------------------------------
extern "C" void kernel_launch(void* const* d_in, const int* in_sizes, int n_in,
                              void* d_out, int out_size, void* d_ws, size_t ws_size,
                              hipStream_t stream)
{
    const float* U   = (const float*)d_in[0];
    const float* K1w = (const float*)d_in[1];
    const float* K2w = (const float*)d_in[2];
    const float* Aww = (const float*)d_in[3];
    float* out = (float*)d_out;
    float* ws  = (float*)d_ws;

    const int nb = in_sizes[0] / (HH * WW);   // batch = 16
    dim3 grid(WW / TILE, HH / TILE, nb);
    dim3 block(256);

    // layer 1: U   -> out
    rd_layer<<<grid, block, 0, stream>>>(U,   out, K1w, K2w, Aww);
    // layer 2: out -> ws
    rd_layer<<<grid, block, 0, stream>>>(out, ws,  K1w, K2w, Aww);
    // layer 3: ws  -> out   (final result lands in d_out)
    rd_layer<<<grid, block, 0, stream>>>(ws,  out, K1w, K2w, Aww);
}